// SpatiallySparseDense_old_2087354106279
// MI455X (gfx1250) — compile-verified
//
#include <hip/hip_runtime.h>
#include <hip/hip_bf16.h>

typedef float v2f __attribute__((ext_vector_type(2)));
typedef float v8f __attribute__((ext_vector_type(8)));

#define N_TOT 32768
#define K_DIM 128
#define B_DIM 8
#define NSEG  32          // N-segments -> 32*8 = 256 workgroups
#define NC    32          // chunk rows per buffer (32 rows x 512B = 16KB per buffer)
#define LSTR  136         // LDS row stride (floats); 2*136 % 64 == 16 -> halves hit disjoint banks

// ---------------------------------------------------------------- zero scratch
__global__ void zero_ws_kernel(float* __restrict__ ws, int n) {
    int i = blockIdx.x * blockDim.x + threadIdx.x;
    if (i < n) ws[i] = 0.0f;
}

// ------------------------------------------------- gram = W0^T W0, m = W0^T d
// Refactored so the LDS tile is RAW mult (async-DMA friendly):
//   gram[k,l] = sum_n (mask^2[n] * mult[n,k]) * mult[n,l]   (scale A side only)
//   m[k]      = sum_n (mask[n]*data[n]) * mult[n,k]
// mult chunks are double-buffered into LDS with GLOBAL_LOAD_ASYNC_TO_LDS_B128
// (ASYNCcnt), overlapped with the 64 V_WMMA_F32_16X16X4_F32 per chunk.
__global__ __launch_bounds__(256) void gram_m_kernel(
    const float* __restrict__ data, const float* __restrict__ mask,
    const float* __restrict__ mult, float* __restrict__ gramW,
    float* __restrict__ mW) {
    __shared__ float ldsRaw[2][NC * LSTR];  // raw mult chunk (async-filled)
    __shared__ float ldsS[2][NC];           // mask^2
    __shared__ float ldsD[2][NC];           // mask*data

    const int t    = threadIdx.x;
    const int b    = blockIdx.y;
    const int seg  = blockIdx.x;
    const int lane = t & 31;
    const int wv   = t >> 5;            // wave id 0..7 -> output row tile
    const int hf   = lane >> 4;         // half-wave: K-pair {0,1} vs {2,3}
    const int lc   = lane & 15;

    const int rowsPerSeg = N_TOT / NSEG;   // 1024
    const int NCH  = rowsPerSeg / NC;      // 32 chunks
    const int nBeg = seg * rowsPerSeg;

    const unsigned lbyte = (unsigned)lane * 16u;   // lane's 16B slice of a row

    // Issue one chunk: each wave DMAs 4 rows (1 x b128 per row) into buffer p,
    // and threads t<NC stage mask^2 / mask*data scalars.
    auto issue_chunk = [&](int c, int p) {
        const int n0 = nBeg + c * NC;
        const unsigned ldsb = (unsigned)(uintptr_t)(&ldsRaw[p][0]);
#pragma unroll
        for (int i = 0; i < 4; ++i) {
            const int row = wv * 4 + i;
            const unsigned dst = ldsb + (unsigned)(row * (LSTR * 4)) + lbyte;
            const unsigned long long src =
                (unsigned long long)(uintptr_t)(mult + (size_t)(n0 + row) * K_DIM);
            asm volatile("global_load_async_to_lds_b128 %0, %1, %2"
                         :: "v"(dst), "v"(lbyte), "s"(src)
                         : "memory");
        }
        if (t < NC) {
            float w = mask[b * N_TOT + n0 + t];
            ldsS[p][t] = w * w;
            ldsD[p][t] = w * data[b * N_TOT + n0 + t];
        }
    };

    v8f zero = {};
    v8f acc[8];
#pragma unroll
    for (int kj = 0; kj < 8; ++kj) acc[kj] = zero;
    float acc_m = 0.0f;

    issue_chunk(0, 0);
    for (int c = 0; c < NCH; ++c) {
        const int p = c & 1;
        if (c + 1 < NCH) {
            issue_chunk(c + 1, 1 - p);               // fill other buffer
            asm volatile("s_wait_asynccnt 0x4" ::: "memory");  // chunk c landed
        } else {
            asm volatile("s_wait_asynccnt 0x0" ::: "memory");
        }
        __syncthreads();   // all waves' DMA + scalar staging for chunk c visible

        const float* raw = &ldsRaw[p][0];
        const float* sb  = &ldsS[p][0];
        const float* db  = &ldsD[p][0];
#pragma unroll
        for (int nn = 0; nn < NC; nn += 4) {
            const int r0 = nn + hf * 2;
            const float rx = raw[r0 * LSTR + wv * 16 + lc];
            const float ry = raw[(r0 + 1) * LSTR + wv * 16 + lc];
            acc_m += rx * db[r0] + ry * db[r0 + 1];
            v2f a;
            a.x = rx * sb[r0];
            a.y = ry * sb[r0 + 1];
#pragma unroll
            for (int kj = 0; kj < 8; ++kj) {
                v2f bv;
                bv.x = raw[r0 * LSTR + kj * 16 + lc];
                bv.y = raw[(r0 + 1) * LSTR + kj * 16 + lc];
                acc[kj] = __builtin_amdgcn_wmma_f32_16x16x4_f32(
                    false, a, false, bv, (short)0, acc[kj], false, false);
            }
        }
        __syncthreads();   // chunk c consumed -> its buffer may be refilled
    }

    // Accumulate partial tiles into global gram / m.
#pragma unroll
    for (int kj = 0; kj < 8; ++kj) {
#pragma unroll
        for (int v = 0; v < 8; ++v) {
            const int row = wv * 16 + hf * 8 + v;   // C/D layout: lanes 16-31 hold M=8..15
            const int col = kj * 16 + lc;
            atomicAdd(&gramW[((size_t)b * K_DIM + row) * K_DIM + col], acc[kj][v]);
        }
    }
    atomicAdd(&mW[b * K_DIM + wv * 16 + lc], acc_m);
}

// ---------------------------------------- solve (gram + sigma^2 I) x = m, per batch
__global__ __launch_bounds__(128) void solve_kernel(
    float* __restrict__ A, float* __restrict__ rhs,
    const float* __restrict__ sigma, float* __restrict__ out) {
    __shared__ float piv[K_DIM];
    __shared__ float fbuf[K_DIM];
    __shared__ float rhsP;

    const int t = threadIdx.x;
    const int b = blockIdx.x;
    float* Ab = A + (size_t)b * K_DIM * K_DIM;
    float* rb = rhs + b * K_DIM;

    Ab[t * K_DIM + t] += sigma[0];
    __threadfence();
    __syncthreads();

    for (int p = 0; p < K_DIM; ++p) {
        piv[t] = Ab[p * K_DIM + t];           // coalesced pivot-row fetch
        __syncthreads();
        const float pr = 1.0f / piv[p];
        const float f  = (t == p) ? 0.0f : Ab[t * K_DIM + p] * pr;
        fbuf[t] = f;
        if (t == p) rhsP = rb[p];
        __syncthreads();
        const float pvt = piv[t];
        for (int r = 0; r < K_DIM; ++r) {     // thread t owns column t: coalesced
            if (r != p) Ab[r * K_DIM + t] -= fbuf[r] * pvt;
        }
        if (t != p) rb[t] -= f * rhsP;
        __threadfence();
        __syncthreads();
    }
    out[b * K_DIM + t] = rb[t] / Ab[t * K_DIM + t];
}

// ---------------------------------------------------------------------- launch
extern "C" void kernel_launch(void* const* d_in, const int* in_sizes, int n_in,
                              void* d_out, int out_size, void* d_ws, size_t ws_size,
                              hipStream_t stream) {
    const float* data  = (const float*)d_in[0];   // [B, 32,32,32]
    const float* mask  = (const float*)d_in[1];   // [B, 32,32,32]
    const float* mult  = (const float*)d_in[2];   // [N, K]
    const float* sigma = (const float*)d_in[3];   // [1]
    float* out = (float*)d_out;                   // [B, K, 1]

    float* gramW = (float*)d_ws;                          // B*K*K floats
    float* mW    = gramW + (size_t)B_DIM * K_DIM * K_DIM; // B*K floats

    const int zn = B_DIM * K_DIM * K_DIM + B_DIM * K_DIM;
    zero_ws_kernel<<<(zn + 255) / 256, 256, 0, stream>>>(gramW, zn);
    gram_m_kernel<<<dim3(NSEG, B_DIM), 256, 0, stream>>>(data, mask, mult, gramW, mW);
    solve_kernel<<<B_DIM, K_DIM, 0, stream>>>(gramW, mW, sigma, out);
}